// HaversineSmoothedLoss_20134806683904
// MI455X (gfx1250) — compile-verified
//
#include <hip/hip_runtime.h>
#include <math.h>

typedef __attribute__((ext_vector_type(2))) float v2f;
typedef __attribute__((ext_vector_type(8))) float v8f;

#define B_ 4096
#define C_ 12288
#define TAU_ 75.0f
#define RADIUS_ 6371.0f
#define EPS_ 1e-9f
#define DEG2RAD 0.017453292519943295f
#define NW 32                       // waves per block (1024 threads)
#define COLS_PER_IT (16 * NW)       // 512 columns per iteration
#define NITER (C_ / COLS_PER_IT)    // 24

__global__ void zero_out_kernel(float* out) {
    if (threadIdx.x == 0) out[0] = 0.0f;
}

// Convert [n,2] (lat,lon) degrees -> unit sphere vectors (x,y,z,0) as float4.
__global__ void prep_points_kernel(const float* __restrict__ ll,
                                   float4* __restrict__ P, int n) {
    int i = blockIdx.x * blockDim.x + threadIdx.x;
    if (i < n) {
        float lat = ll[2 * i]     * DEG2RAD;
        float lon = ll[2 * i + 1] * DEG2RAD;
        float cl = cosf(lat), sl = sinf(lat);
        P[i] = make_float4(cl * cosf(lon), cl * sinf(lon), sl, 0.0f);
    }
}

// Branchless asin for aa = x^2, x = sqrt(aa) in [0,1] (Cephes minimax, ~1e-7).
// Returns asin(sqrt(aa)). Uses raw v_sqrt_f32 (domain is safe, no denorm care).
__device__ __forceinline__ float asin_sqrt_fast(float aa) {
    float s   = __builtin_amdgcn_sqrtf(aa);       // sin(c/2)
    bool  big = s > 0.5f;
    float z   = big ? 0.5f * (1.0f - s) : aa;     // reduced poly argument
    float arg = __builtin_amdgcn_sqrtf(z);        // == s when !big
    float p = 4.2163199048e-2f;
    p = fmaf(p, z, 2.4181311049e-2f);
    p = fmaf(p, z, 4.5470025998e-2f);
    p = fmaf(p, z, 7.4953002686e-2f);
    p = fmaf(p, z, 1.6666752422e-1f);
    float r = fmaf(p * z, arg, arg);              // asin(arg-domain)
    return big ? fmaf(-2.0f, r, 1.5707963268f) : r;
}

// One block = 16 rows x all C. 32 waves; each wave does 16x16 WMMA tiles.
__launch_bounds__(1024)
__global__ void hav_loss_kernel(const float* __restrict__ logits,
                                const float4* __restrict__ P,
                                const float4* __restrict__ Q,
                                float* __restrict__ out) {
    __shared__ float ls_m[NW][16], ls_z[NW][16], ls_s[NW][16], ls_l[NW][16];
    __shared__ float fm[16], fzinv[16], fsinv[16];

    const int tid  = threadIdx.x;
    const int wave = tid >> 5;
    const int lane = tid & 31;
    const int half = lane >> 4;   // K-half for A/B operands, row-half for D
    const int l16  = lane & 15;
    const int rowBase = blockIdx.x * 16;

    // A operand (16x4 f32): lanes 0-15 carry K=0,1; lanes 16-31 carry K=2,3.
    float4 p = P[rowBase + l16];
    v2f A;
    A.x = half ? p.z : p.x;
    A.y = half ? p.w : p.y;

    float mx[8], zz[8], sw[8];
#pragma unroll
    for (int r = 0; r < 8; ++r) { mx[r] = -3.0e38f; zz[r] = 0.0f; sw[r] = 0.0f; }

    const float kDistScale = -(2.0f * RADIUS_) / TAU_;  // exp(kDistScale*asin(s))

    // ---------------- Phase 1: row sum of w, online softmax stats ------------
    for (int it = 0; it < NITER; ++it) {
        int col = it * COLS_PER_IT + wave * 16 + l16;
        float4 q = Q[col];
        v2f Bv;
        Bv.x = half ? q.z : q.x;
        Bv.y = half ? q.w : q.y;
        v8f cz = {};
        v8f d = __builtin_amdgcn_wmma_f32_16x16x4_f32(
            false, A, false, Bv, (short)0, cz, false, false);
#pragma unroll
        for (int r = 0; r < 8; ++r) {
            int row = rowBase + r + 8 * half;
            float aa = 0.5f * (1.0f - d[r]);
            aa = fminf(fmaxf(aa, 0.0f), 1.0f);
            float w = __expf(kDistScale * asin_sqrt_fast(aa));
            sw[r] += w;
            float x = logits[row * C_ + col];
            float mo = mx[r];
            float mn = fmaxf(mo, x);
            zz[r] = zz[r] * __expf(mo - mn) + __expf(x - mn);
            mx[r] = mn;
        }
    }

    // Reduce across the 16 lanes sharing each row (wave32: two 16-groups).
#pragma unroll
    for (int r = 0; r < 8; ++r) {
#pragma unroll
        for (int off = 8; off >= 1; off >>= 1) {
            float mo = __shfl_xor(mx[r], off, 16);
            float zo = __shfl_xor(zz[r], off, 16);
            float so = __shfl_xor(sw[r], off, 16);
            float mn = fmaxf(mx[r], mo);
            zz[r] = zz[r] * __expf(mx[r] - mn) + zo * __expf(mo - mn);
            mx[r] = mn;
            sw[r] += so;
        }
    }
    if (l16 == 0) {
#pragma unroll
        for (int r = 0; r < 8; ++r) {
            int row = r + 8 * half;
            ls_m[wave][row] = mx[r];
            ls_z[wave][row] = zz[r];
            ls_s[wave][row] = sw[r];
        }
    }
    __syncthreads();

    // Merge the NW waves' column partitions; thread t owns row t.
    if (tid < 16) {
        float M = -3.0e38f, Z = 0.0f, S = 0.0f;
        for (int w2 = 0; w2 < NW; ++w2) {
            float mo = ls_m[w2][tid], zo = ls_z[w2][tid];
            float mn = fmaxf(M, mo);
            Z = Z * __expf(M - mn) + zo * __expf(mo - mn);
            M = mn;
            S += ls_s[w2][tid];
        }
        fm[tid]    = M;
        fzinv[tid] = __frcp_rn(Z);
        fsinv[tid] = __frcp_rn(S);
    }
    __syncthreads();

    float rm[8], rz[8], acc[8];
#pragma unroll
    for (int r = 0; r < 8; ++r) {
        int row = r + 8 * half;
        rm[r] = fm[row];
        rz[r] = fzinv[row];
        acc[r] = 0.0f;
    }

    // ---------------- Phase 2: accumulate sum_c w * log(p + eps) ------------
    for (int it = 0; it < NITER; ++it) {
        int col = it * COLS_PER_IT + wave * 16 + l16;
        float4 q = Q[col];
        v2f Bv;
        Bv.x = half ? q.z : q.x;
        Bv.y = half ? q.w : q.y;
        v8f cz = {};
        v8f d = __builtin_amdgcn_wmma_f32_16x16x4_f32(
            false, A, false, Bv, (short)0, cz, false, false);
#pragma unroll
        for (int r = 0; r < 8; ++r) {
            int row = rowBase + r + 8 * half;
            float aa = 0.5f * (1.0f - d[r]);
            aa = fminf(fmaxf(aa, 0.0f), 1.0f);
            float w = __expf(kDistScale * asin_sqrt_fast(aa));
            float x = logits[row * C_ + col];
            float pprob = __expf(x - rm[r]) * rz[r];
            acc[r] += w * __logf(pprob + EPS_);
        }
    }

#pragma unroll
    for (int r = 0; r < 8; ++r) {
#pragma unroll
        for (int off = 8; off >= 1; off >>= 1)
            acc[r] += __shfl_xor(acc[r], off, 16);
    }
    if (l16 == 0) {
#pragma unroll
        for (int r = 0; r < 8; ++r) ls_l[wave][r + 8 * half] = acc[r];
    }
    __syncthreads();

    if (tid < 16) {
        float s = 0.0f;
        for (int w2 = 0; w2 < NW; ++w2) s += ls_l[w2][tid];
        float row_loss = -s * fsinv[tid];   // -(1/Sw) * sum w*log(p+eps)
#pragma unroll
        for (int off = 8; off >= 1; off >>= 1)
            row_loss += __shfl_xor(row_loss, off, 16);
        if (tid == 0) atomicAdd(out, row_loss * (1.0f / (float)B_));
    }
}

extern "C" void kernel_launch(void* const* d_in, const int* in_sizes, int n_in,
                              void* d_out, int out_size, void* d_ws, size_t ws_size,
                              hipStream_t stream) {
    const float* logits = (const float*)d_in[0];   // [B,C] f32
    const float* latlon = (const float*)d_in[1];   // [B,2] f32
    const float* cents  = (const float*)d_in[2];   // [C,2] f32
    // d_in[3] (geocell_indices) is unused by the reference loss.
    float* out = (float*)d_out;

    float4* P = (float4*)d_ws;                               // B*16 bytes
    float4* Q = (float4*)((char*)d_ws + (size_t)B_ * 16);    // C*16 bytes

    zero_out_kernel<<<1, 32, 0, stream>>>(out);
    prep_points_kernel<<<(B_ + 255) / 256, 256, 0, stream>>>(latlon, P, B_);
    prep_points_kernel<<<(C_ + 255) / 256, 256, 0, stream>>>(cents, Q, C_);
    hav_loss_kernel<<<B_ / 16, 1024, 0, stream>>>(logits, P, Q, out);
}